// Mahony_61048665145725
// MI455X (gfx1250) — compile-verified
//
#include <hip/hip_runtime.h>
#include <stdint.h>

// ---------------- configuration ----------------
#define CHUNK_ROWS     512
#define CHUNK_FLOATS   (CHUNK_ROWS * 6)        // 3072 floats = 12 KB per buffer
#define VEC4_PER_CHUNK (CHUNK_FLOATS / 4)      // 768 float4 slots
#define OPS_PER_CHUNK  (VEC4_PER_CHUNK / 32)   // 24 async b128 issues per chunk (wave32)

typedef int v4i_t __attribute__((vector_size(16)));
typedef __attribute__((address_space(1))) v4i_t gbl_v4i_t;
typedef __attribute__((address_space(3))) v4i_t lds_v4i_t;

// ---- CDNA5 async global->LDS copy (16B per lane per issue) ----
__device__ __forceinline__ void async_b128(const float* gsrc, const float* ldst) {
#if __has_builtin(__builtin_amdgcn_global_load_async_to_lds_b128)
  __builtin_amdgcn_global_load_async_to_lds_b128(
      (gbl_v4i_t*)(uintptr_t)gsrc,
      (lds_v4i_t*)(uintptr_t)(uint32_t)(uintptr_t)ldst,
      0, 0);
#else
  unsigned l = (unsigned)(uintptr_t)ldst;   // generic LDS ptr low 32 bits = LDS offset
  asm volatile("global_load_async_to_lds_b128 %0, %1, off"
               :: "v"(l), "v"(gsrc) : "memory");
#endif
}

__device__ __forceinline__ void wait_async_prev() {  // ASYNCcnt <= OPS_PER_CHUNK
#if __has_builtin(__builtin_amdgcn_s_wait_asynccnt)
  __builtin_amdgcn_s_wait_asynccnt(OPS_PER_CHUNK);
#else
  asm volatile("s_wait_asynccnt 24" ::: "memory");
#endif
  asm volatile("" ::: "memory");
}

__device__ __forceinline__ void wait_async_all() {   // ASYNCcnt == 0
#if __has_builtin(__builtin_amdgcn_s_wait_asynccnt)
  __builtin_amdgcn_s_wait_asynccnt(0);
#else
  asm volatile("s_wait_asynccnt 0" ::: "memory");
#endif
  asm volatile("" ::: "memory");
}

__device__ __forceinline__ void issue_chunk(const float* gsrc, const float* lbuf, int lane) {
#pragma unroll
  for (int i = 0; i < OPS_PER_CHUNK; ++i) {
    const int idx4 = i * 32 + lane;                  // float4 index within chunk
    async_b128(gsrc + (size_t)idx4 * 4, lbuf + (size_t)idx4 * 4);
  }
}

// ---- one Mahony step (branch-free; ki<0 folded into iflag) ----
__device__ __forceinline__ void mahony_step(
    float ax, float ay, float az, float gx, float gy, float gz,
    float& q0, float& q1, float& q2, float& q3,
    float& i0, float& i1, float& i2,
    float kp, float kis, float iflag)
{
  // normalize accel
  const float rn = __builtin_amdgcn_rsqf(ax * ax + ay * ay + az * az);
  ax *= rn; ay *= rn; az *= rn;

  // half gravity direction from quaternion
  const float hv0 = q1 * q3 - q0 * q2;
  const float hv1 = q0 * q1 + q2 * q3;
  const float hv2 = q0 * q0 + q3 * q3 - 0.5f;

  // halfe = a x halfv
  const float he0 = ay * hv2 - az * hv1;
  const float he1 = az * hv0 - ax * hv2;
  const float he2 = ax * hv1 - ay * hv0;

  // integrator: if ki<0 it is forced to zero every step -> iflag = 0
  i0 = iflag * (i0 + he0);
  i1 = iflag * (i1 + he1);
  i2 = iflag * (i2 + he2);

  gx += kp * he0 + kis * i0;
  gy += kp * he1 + kis * i1;
  gz += kp * he2 + kis * i2;

  // quaternion update (reference's exact form)
  const float n0 = q0 - gx * q1 - gy * q2 - gz * q3;
  const float n1 = q1 + q0 * gx + q2 * gz - q3 * gy;
  const float n2 = q2 + q0 * gy - q1 * gz + q3 * gx;
  const float n3 = q3 + q0 * gz + q1 * gy - q2 * gx;

  const float rq = __builtin_amdgcn_rsqf(n0 * n0 + n1 * n1 + n2 * n2 + n3 * n3);
  q0 = n0 * rq; q1 = n1 * rq; q2 = n2 * rq; q3 = n3 * rq;
}

__global__ __launch_bounds__(32) void mahony_scan_kernel(
    const float* __restrict__ q_init,
    const float* __restrict__ imu,
    const float* __restrict__ w1,
    const float* __restrict__ w2,
    float* __restrict__ out,
    int T)
{
  __shared__ __align__(16) float lds[2][CHUNK_FLOATS];
  const int lane = threadIdx.x;

  const float kp = w1[0];
  const float ki = w2[0];
  const float kis = ki * (1.0f / 50.0f);
  const float iflag = (ki < 0.0f) ? 0.0f : 1.0f;

  const float qi = q_init[0];
  float q0 = qi, q1 = qi, q2 = qi, q3 = qi;
  float i0 = 0.0f, i1 = 0.0f, i2 = 0.0f;

  const int nChunks = T / CHUNK_ROWS;

  if (nChunks > 0) {
    issue_chunk(imu, &lds[0][0], lane);  // prime the pipeline
  }

  for (int c = 0; c < nChunks; ++c) {
    const int nb = c & 1;
    if (c + 1 < nChunks) {
      // start next chunk into the other buffer, then wait only for chunk c
      issue_chunk(imu + (size_t)(c + 1) * CHUNK_FLOATS, &lds[nb ^ 1][0], lane);
      wait_async_prev();
    } else {
      wait_async_all();
    }

    const float2* rp = (const float2*)&lds[nb][0];   // rows are 24B -> 8B aligned
#pragma unroll 4
    for (int r = 0; r < CHUNK_ROWS; ++r) {
      const float2 v0 = rp[r * 3 + 0];
      const float2 v1 = rp[r * 3 + 1];
      const float2 v2 = rp[r * 3 + 2];
      mahony_step(v0.x, v0.y, v1.x, v1.y, v2.x, v2.y,
                  q0, q1, q2, q3, i0, i1, i2, kp, kis, iflag);
    }
  }

  // tail rows straight from global (T mod CHUNK_ROWS, <= 511 rows)
  for (int r = nChunks * CHUNK_ROWS; r < T; ++r) {
    const float2* rp = (const float2*)(imu + (size_t)r * 6);
    const float2 v0 = rp[0];
    const float2 v1 = rp[1];
    const float2 v2 = rp[2];
    mahony_step(v0.x, v0.y, v1.x, v1.y, v2.x, v2.y,
                q0, q1, q2, q3, i0, i1, i2, kp, kis, iflag);
  }

  if (lane == 0) {
    out[0] = q0;
    out[1] = q1;
    out[2] = q2;
    out[3] = q3;
  }
}

extern "C" void kernel_launch(void* const* d_in, const int* in_sizes, int n_in,
                              void* d_out, int out_size, void* d_ws, size_t ws_size,
                              hipStream_t stream) {
  const float* q_init = (const float*)d_in[0];
  const float* imu    = (const float*)d_in[1];
  const float* w1     = (const float*)d_in[2];
  const float* w2     = (const float*)d_in[3];
  float* out = (float*)d_out;

  const int T = in_sizes[1] / 6;

  hipLaunchKernelGGL(mahony_scan_kernel, dim3(1), dim3(32), 0, stream,
                     q_init, imu, w1, w2, out, T);
}